// TreeContextEncoder_28329604284737
// MI455X (gfx1250) — compile-verified
//
#include <hip/hip_runtime.h>
#include <math.h>

#define N_NODES 300000
#define N_EDGES 600000
#define IN_DIM  64
#define HID     128
#define LAYERS  2
#define BN_EPS  1e-5f
#define INV_SQRT_H 0.08838834764831845f  // 1/sqrt(128)

typedef float v2f __attribute__((ext_vector_type(2)));
typedef float v8f __attribute__((ext_vector_type(8)));

// CDNA5 fp32 WMMA: D(16x16) = A(16x4) * B(4x16) + C
__device__ __forceinline__ v8f wmma_f32_16x16x4(v2f a, v2f b, v8f c) {
    return __builtin_amdgcn_wmma_f32_16x16x4_f32(
        /*neg_a=*/false, a, /*neg_b=*/false, b,
        /*c_mod=*/(short)0, c, /*reuse_a=*/false, /*reuse_b=*/false);
}

// ---------------------------------------------------------------------------
// h = X @ W_in + b_in        X:[N,64]  W:[64,128]  H:[N,128]
// Block = 256 thr = 8 waves. Block -> 16 output rows; wave w -> cols [16w,16w+16)
// ---------------------------------------------------------------------------
__global__ __launch_bounds__(256)
void gemm_in_kernel(const float* __restrict__ X, const float* __restrict__ W,
                    const float* __restrict__ bias, float* __restrict__ H)
{
    __shared__ float As[16][IN_DIM + 4];
    const int tid  = threadIdx.x;
    const int row0 = blockIdx.x * 16;

    // cooperative load of 16x64 A tile (1024 floats, 4 per thread)
    {
        const int idx = tid * 4;
        const int r = idx / IN_DIM;
        const int c = idx % IN_DIM;
        const float4 v = *(const float4*)(X + (size_t)(row0 + r) * IN_DIM + c);
        As[r][c + 0] = v.x; As[r][c + 1] = v.y;
        As[r][c + 2] = v.z; As[r][c + 3] = v.w;
    }
    __syncthreads();

    const int wave   = tid >> 5;
    const int lane   = tid & 31;
    const int lane15 = lane & 15;
    const int kOff   = (lane < 16) ? 0 : 2;   // ISA A-frag: lanes16-31 hold K=2,3
    const int col    = wave * 16 + lane15;

    v8f acc = {};
    #pragma unroll
    for (int k = 0; k < IN_DIM; k += 4) {
        v2f a, b;
        a.x = As[lane15][k + kOff + 0];
        a.y = As[lane15][k + kOff + 1];
        b.x = W[(size_t)(k + kOff + 0) * HID + col];
        b.y = W[(size_t)(k + kOff + 1) * HID + col];
        acc = wmma_f32_16x16x4(a, b, acc);
    }

    const float bv    = bias[col];
    const int   rbase = row0 + ((lane < 16) ? 0 : 8);   // C/D frag row mapping
    #pragma unroll
    for (int r = 0; r < 8; ++r)
        H[(size_t)(rbase + r) * HID + col] = acc[r] + bv;
}

// ---------------------------------------------------------------------------
// Edge pass: w_e = exp(dot(h[src],att_k)/sqrt(H)) / (1+len); agg[dst] += w_e*h[src]
// One wave (32 lanes x float4 = 128 ch) per edge; 8 edges per block.
// ---------------------------------------------------------------------------
__global__ __launch_bounds__(256)
void edge_kernel(const float* __restrict__ H, const int* __restrict__ src,
                 const int* __restrict__ dst, const float* __restrict__ elen,
                 const float* __restrict__ att_k, float* __restrict__ agg)
{
    const int wave = threadIdx.x >> 5;
    const int lane = threadIdx.x & 31;
    const int e    = blockIdx.x * 8 + wave;

    const int s = src[e];
    const int d = dst[e];

    const float4 hv = ((const float4*)(H + (size_t)s * HID))[lane];
    const float4 ak = ((const float4*)att_k)[lane];

    float t = hv.x * ak.x + hv.y * ak.y + hv.z * ak.z + hv.w * ak.w;
    #pragma unroll
    for (int off = 16; off > 0; off >>= 1)      // wave32 butterfly reduce
        t += __shfl_xor(t, off, 32);

    const float w = __expf(t * INV_SQRT_H) / (1.0f + elen[e]);

    float* out = agg + (size_t)d * HID + lane * 4;
    atomicAdd(out + 0, w * hv.x);
    atomicAdd(out + 1, w * hv.y);
    atomicAdd(out + 2, w * hv.z);
    atomicAdd(out + 3, w * hv.w);
}

// ---------------------------------------------------------------------------
// u = h + relu(agg) @ Wl + bl ; also accumulate per-channel sum / sum^2 for BN.
// ---------------------------------------------------------------------------
__global__ __launch_bounds__(256)
void layer_gemm_kernel(const float* __restrict__ Hin, const float* __restrict__ agg,
                       const float* __restrict__ W,   const float* __restrict__ bias,
                       float* __restrict__ U, float* __restrict__ sum,
                       float* __restrict__ sumsq)
{
    __shared__ float As[16][HID + 4];
    const int tid  = threadIdx.x;
    const int row0 = blockIdx.x * 16;

    // stage relu(agg) 16x128 tile (2048 floats, 8 per thread)
    {
        const int idx = tid * 8;
        const int r = idx / HID;
        const int c = idx % HID;
        const float4 v0 = *(const float4*)(agg + (size_t)(row0 + r) * HID + c);
        const float4 v1 = *(const float4*)(agg + (size_t)(row0 + r) * HID + c + 4);
        As[r][c + 0] = fmaxf(v0.x, 0.f); As[r][c + 1] = fmaxf(v0.y, 0.f);
        As[r][c + 2] = fmaxf(v0.z, 0.f); As[r][c + 3] = fmaxf(v0.w, 0.f);
        As[r][c + 4] = fmaxf(v1.x, 0.f); As[r][c + 5] = fmaxf(v1.y, 0.f);
        As[r][c + 6] = fmaxf(v1.z, 0.f); As[r][c + 7] = fmaxf(v1.w, 0.f);
    }
    __syncthreads();

    const int wave   = tid >> 5;
    const int lane   = tid & 31;
    const int lane15 = lane & 15;
    const int kOff   = (lane < 16) ? 0 : 2;
    const int col    = wave * 16 + lane15;

    v8f acc = {};
    #pragma unroll
    for (int k = 0; k < HID; k += 4) {
        v2f a, b;
        a.x = As[lane15][k + kOff + 0];
        a.y = As[lane15][k + kOff + 1];
        b.x = W[(size_t)(k + kOff + 0) * HID + col];
        b.y = W[(size_t)(k + kOff + 1) * HID + col];
        acc = wmma_f32_16x16x4(a, b, acc);
    }

    const float bv    = bias[col];
    const int   rbase = row0 + ((lane < 16) ? 0 : 8);

    float s = 0.f, s2 = 0.f;
    #pragma unroll
    for (int r = 0; r < 8; ++r) {
        const size_t o = (size_t)(rbase + r) * HID + col;
        const float u = Hin[o] + acc[r] + bv;
        U[o] = u;
        s  += u;
        s2 += u * u;
    }
    // lanes l and l+16 cover the two row-halves of the same column: combine
    s  += __shfl_xor(s, 16, 32);
    s2 += __shfl_xor(s2, 16, 32);
    if (lane < 16) {
        atomicAdd(&sum[col],   s);
        atomicAdd(&sumsq[col], s2);
    }
}

// ---------------------------------------------------------------------------
// Fold BN stats + gamma/beta into per-channel scale/shift.
// ---------------------------------------------------------------------------
__global__ void bn_finalize_kernel(const float* __restrict__ sum,
                                   const float* __restrict__ sumsq,
                                   const float* __restrict__ gamma,
                                   const float* __restrict__ beta,
                                   float* __restrict__ scale,
                                   float* __restrict__ shift)
{
    const int c = threadIdx.x;            // 128 threads
    const float invN = 1.0f / (float)N_NODES;
    const float mean = sum[c] * invN;
    const float var  = fmaxf(sumsq[c] * invN - mean * mean, 0.0f);
    const float sc   = gamma[c] * rsqrtf(var + BN_EPS);
    scale[c] = sc;
    shift[c] = beta[c] - mean * sc;
}

// ---------------------------------------------------------------------------
// h_out = u * scale[c] + shift[c]   (float4 streaming pass)
// ---------------------------------------------------------------------------
__global__ __launch_bounds__(256)
void bn_apply_kernel(const float* __restrict__ U, const float* __restrict__ scale,
                     const float* __restrict__ shift, float* __restrict__ Hout)
{
    const size_t i = ((size_t)blockIdx.x * 256 + threadIdx.x) * 4;
    const int c = (int)(i & (HID - 1));
    const float4 u = *(const float4*)(U + i);
    float4 o;
    o.x = u.x * scale[c + 0] + shift[c + 0];
    o.y = u.y * scale[c + 1] + shift[c + 1];
    o.z = u.z * scale[c + 2] + shift[c + 2];
    o.w = u.w * scale[c + 3] + shift[c + 3];
    *(float4*)(Hout + i) = o;
}

__global__ __launch_bounds__(256)
void fill_zero_kernel(float* __restrict__ p, size_t n)
{
    const size_t i = (size_t)blockIdx.x * blockDim.x + threadIdx.x;
    if (i < n) p[i] = 0.0f;
}

// ---------------------------------------------------------------------------
extern "C" void kernel_launch(void* const* d_in, const int* in_sizes, int n_in,
                              void* d_out, int out_size, void* d_ws, size_t ws_size,
                              hipStream_t stream)
{
    const float* node_init = (const float*)d_in[0];
    const int*   edge_idx  = (const int*)  d_in[1];   // [2,E] row-major (int32 under jax default)
    const float* edge_len  = (const float*)d_in[2];
    const float* W_in      = (const float*)d_in[3];
    const float* b_in      = (const float*)d_in[4];
    const float* att_k     = (const float*)d_in[5];
    const float* Wl        = (const float*)d_in[6];   // [2,128,128]
    const float* bl        = (const float*)d_in[7];   // [2,128]
    const float* gamma     = (const float*)d_in[8];
    const float* beta      = (const float*)d_in[9];

    const int* src = edge_idx;
    const int* dst = edge_idx + N_EDGES;

    const size_t NH = (size_t)N_NODES * HID;
    float* hA    = (float*)d_out;              // ping buffer; final result ends here
    float* agg   = (float*)d_ws;               // [N,H]
    float* hB    = agg + NH;                   // [N,H] pre-BN u / pong buffer
    float* stats = hB + NH;                    // 512 floats
    float* sum   = stats;
    float* sumsq = stats + 128;
    float* scale = stats + 256;
    float* shift = stats + 384;

    // h = X @ W_in + b_in   (N divisible by 16: 18750 tiles)
    gemm_in_kernel<<<N_NODES / 16, 256, 0, stream>>>(node_init, W_in, b_in, hA);

    for (int l = 0; l < LAYERS; ++l) {
        fill_zero_kernel<<<(int)((NH + 255) / 256), 256, 0, stream>>>(agg, NH);
        fill_zero_kernel<<<1, 256, 0, stream>>>(stats, 256);

        edge_kernel<<<N_EDGES / 8, 256, 0, stream>>>(hA, src, dst, edge_len, att_k, agg);

        layer_gemm_kernel<<<N_NODES / 16, 256, 0, stream>>>(
            hA, agg, Wl + (size_t)l * HID * HID, bl + (size_t)l * HID,
            hB, sum, sumsq);

        bn_finalize_kernel<<<1, 128, 0, stream>>>(
            sum, sumsq, gamma + (size_t)l * HID, beta + (size_t)l * HID, scale, shift);

        bn_apply_kernel<<<(int)(NH / (256 * 4)), 256, 0, stream>>>(hB, scale, shift, hA);
    }
}